// SSMCore_16415365005548
// MI455X (gfx1250) — compile-verified
//
#include <hip/hip_runtime.h>
#include <math.h>

// ---- problem constants (match reference) ----
#define BSZ 2
#define TLEN 2048
#define DM   2048
#define NST  16
#define RR   128
#define NCOL 160   // R + 2N

typedef __attribute__((ext_vector_type(2))) float v2f;
typedef __attribute__((ext_vector_type(8))) float v8f;

// =====================================================================
// Kernel 1: xz(M=4096,160) = x(4096,2048) @ W_xproj(160,2048)^T
// f32 WMMA 16x16x4, one wave per 16x16 output tile.
// A-lane layout (ISA 7.12.2, 32-bit A 16x4): lane L holds row M=L%16,
// K = k0 + 2*(L/16) .. +1 as a float2.  B mirrors per column.
// =====================================================================
__global__ __launch_bounds__(32)
void xproj_wmma(const float* __restrict__ x, const float* __restrict__ Wx,
                float* __restrict__ xz) {
    const int tile = blockIdx.x;          // 256 row-tiles * 10 col-tiles
    const int mt = tile / 10;
    const int nt = tile % 10;
    const int lane = threadIdx.x;
    const int half = lane >> 4;
    const int lr   = lane & 15;

    const float* arow = x  + (size_t)(mt * 16 + lr) * DM + 2 * half;
    const float* brow = Wx + (size_t)(nt * 16 + lr) * DM + 2 * half;

    v8f c0 = {}, c1 = {};
#pragma unroll 4
    for (int k = 0; k < DM; k += 8) {
        v2f a0 = *(const v2f*)(arow + k);
        v2f b0 = *(const v2f*)(brow + k);
        c0 = __builtin_amdgcn_wmma_f32_16x16x4_f32(false, a0, false, b0,
                                                   (short)0, c0, false, false);
        v2f a1 = *(const v2f*)(arow + k + 4);
        v2f b1 = *(const v2f*)(brow + k + 4);
        c1 = __builtin_amdgcn_wmma_f32_16x16x4_f32(false, a1, false, b1,
                                                   (short)0, c1, false, false);
    }
#pragma unroll
    for (int v = 0; v < 8; ++v) {
        // C/D layout: VGPR v holds (M = v + 8*half, N = lr)
        const int row = mt * 16 + v + half * 8;
        const int col = nt * 16 + lr;
        xz[(size_t)row * NCOL + col] = c0[v] + c1[v];
    }
}

// =====================================================================
// Kernel 2: delta(4096,2048) = softplus( xz[:, :128] @ W_dt(2048,128)^T + b_dt )
// Same f32 WMMA tiling, K = 128 (32 wmma steps, 2 interleaved accumulators).
// =====================================================================
__global__ __launch_bounds__(32)
void dt_wmma(const float* __restrict__ xz, const float* __restrict__ Wdt,
             const float* __restrict__ bdt, float* __restrict__ delta) {
    const int tile = blockIdx.x;          // 256 row-tiles * 128 col-tiles
    const int mt = tile >> 7;
    const int nt = tile & 127;
    const int lane = threadIdx.x;
    const int half = lane >> 4;
    const int lr   = lane & 15;

    const float* arow = xz  + (size_t)(mt * 16 + lr) * NCOL + 2 * half; // cols 0..127 = delta_r
    const float* brow = Wdt + (size_t)(nt * 16 + lr) * RR   + 2 * half;

    v8f c0 = {}, c1 = {};
#pragma unroll
    for (int k = 0; k < RR; k += 8) {
        v2f a0 = *(const v2f*)(arow + k);
        v2f b0 = *(const v2f*)(brow + k);
        c0 = __builtin_amdgcn_wmma_f32_16x16x4_f32(false, a0, false, b0,
                                                   (short)0, c0, false, false);
        v2f a1 = *(const v2f*)(arow + k + 4);
        v2f b1 = *(const v2f*)(brow + k + 4);
        c1 = __builtin_amdgcn_wmma_f32_16x16x4_f32(false, a1, false, b1,
                                                   (short)0, c1, false, false);
    }
#pragma unroll
    for (int v = 0; v < 8; ++v) {
        const int row = mt * 16 + v + half * 8;
        const int col = nt * 16 + lr;
        float s = c0[v] + c1[v] + bdt[col];
        // softplus, numerically safe
        delta[(size_t)row * DM + col] = (s > 20.f) ? s : log1pf(__expf(s));
    }
}

// =====================================================================
// Kernel 3: fused selective scan.
// One thread per (b, d, n): 65,536 threads = 2048 wave32s.
// h_{t} = exp(delta*A[d,n]) * h_{t-1} + delta * B[t,n] * x ; y = sum_n h*C[t,n]
// B/C (32 floats per t, shared across all d) staged through LDS in chunks.
// Sum over n via 4 shfl_xor steps inside 16-lane groups.
// =====================================================================
#define TCHUNK 64

__global__ __launch_bounds__(256)
void scan_kernel(const float* __restrict__ x, const float* __restrict__ xz,
                 const float* __restrict__ delta, const float* __restrict__ A_log,
                 const float* __restrict__ Dp, float* __restrict__ out) {
    __shared__ float sBC[TCHUNK * 32];    // [tt][0..15]=B, [tt][16..31]=C

    const int tid  = threadIdx.x;
    const int n    = tid & 15;            // state index
    const int dloc = tid >> 4;            // 0..15 local channel
    const int blk  = blockIdx.x;          // 2 * 128 blocks
    const int b    = blk >> 7;
    const int dblk = blk & 127;
    const int d    = dblk * 16 + dloc;
    const size_t bT = (size_t)b * TLEN;

    const float An = -__expf(A_log[(size_t)d * NST + n]);  // A = -exp(A_log)
    const float dp = Dp[d];
    float h = 0.f;

    for (int t0 = 0; t0 < TLEN; t0 += TCHUNK) {
        // cooperative stage of B/C columns (xz cols 128..159) for TCHUNK steps
#pragma unroll
        for (int j = 0; j < (TCHUNK * 32) / 256; ++j) {
            const int idx = j * 256 + tid;
            const int tt = idx >> 5, cc = idx & 31;
            sBC[idx] = xz[(bT + t0 + tt) * NCOL + RR + cc];
        }
        __syncthreads();

        if (t0 + TCHUNK < TLEN) {   // gfx1250 global_prefetch_b8 for next chunk
            __builtin_prefetch(delta + (bT + t0 + TCHUNK) * DM + d, 0, 3);
            __builtin_prefetch(x     + (bT + t0 + TCHUNK) * DM + d, 0, 3);
        }

        for (int tt = 0; tt < TCHUNK; ++tt) {
            const size_t row = bT + t0 + tt;
            const float dl = delta[row * DM + d];   // broadcast across 16 lanes
            const float xv = x[row * DM + d];
            const float Bv = sBC[tt * 32 + n];      // LDS broadcast reads
            const float Cv = sBC[tt * 32 + 16 + n];

            const float dA = __expf(dl * An);
            h = fmaf(dA, h, dl * Bv * xv);

            float y = h * Cv;                        // reduce over n (16 lanes)
            y += __shfl_xor(y, 1, 16);
            y += __shfl_xor(y, 2, 16);
            y += __shfl_xor(y, 4, 16);
            y += __shfl_xor(y, 8, 16);
            if (n == 0) out[row * DM + d] = y + xv * dp;
        }
        __syncthreads();
    }
}

// =====================================================================
extern "C" void kernel_launch(void* const* d_in, const int* in_sizes, int n_in,
                              void* d_out, int out_size, void* d_ws, size_t ws_size,
                              hipStream_t stream) {
    const float* x    = (const float*)d_in[0];   // (B,T,D)
    const float* Wx   = (const float*)d_in[1];   // (160, D)
    const float* Wdt  = (const float*)d_in[2];   // (D, R)
    const float* bdt  = (const float*)d_in[3];   // (D,)
    const float* Alog = (const float*)d_in[4];   // (D, N)
    const float* Dp   = (const float*)d_in[5];   // (D,)
    float* out = (float*)d_out;                  // (B,T,D)

    float* xz    = (float*)d_ws;                          // 4096*160 f32
    float* delta = xz + (size_t)(BSZ * TLEN) * NCOL;      // 4096*2048 f32

    // GEMM 1: 256 x 10 tiles of 16x16, one wave each
    xproj_wmma<<<256 * 10, 32, 0, stream>>>(x, Wx, xz);
    // GEMM 2 + softplus: 256 x 128 tiles
    dt_wmma<<<256 * 128, 32, 0, stream>>>(xz, Wdt, bdt, delta);
    // fused scan: 2 * 128 blocks of 256 threads (one thread per (b,d,n))
    scan_kernel<<<BSZ * 128, 256, 0, stream>>>(x, xz, delta, Alog, Dp, out);
}